// WLMLayer_28741921145086
// MI455X (gfx1250) — compile-verified
//
#include <hip/hip_runtime.h>
#include <cstddef>
#include <cstdint>

typedef float v2f __attribute__((ext_vector_type(2)));
typedef float v8f __attribute__((ext_vector_type(8)));
typedef unsigned int u32x4 __attribute__((ext_vector_type(4)));
typedef int i32x4 __attribute__((ext_vector_type(4)));
typedef int i32x8 __attribute__((ext_vector_type(8)));

constexpr int kB = 8;
constexpr int kS = 4096;
constexpr int kE = 1024;
constexpr int kTS = 128;      // output times per tile
constexpr int kXOFF = 18;     // x halo before tile start
constexpr int kXUSE = 174;    // x samples actually needed
constexpr int kXWIN = 176;    // padded
constexpr int kXSTR = 17;     // LDS row stride (bank-conflict pad; TDM pad reproduces it)
constexpr int kNC = 80;       // coeff positions stored (rel m in [-4,76))
constexpr int kCFOFF = 4;

// LDS float offsets
constexpr int kOffXS   = 0;
constexpr int kOffCF   = kOffXS + kXWIN * kXSTR;        // 2992
constexpr int kOffFT   = kOffCF + kNC * 8 * kXSTR;      // 13872
constexpr int kOffGP   = kOffFT + 256;                  // 14128
constexpr int kOffPRM  = kOffGP + 8 * 48;               // 14512
constexpr int kOffWT   = kOffPRM + 48;                  // 14560
constexpr int kSmemF   = kOffWT + 4;                    // 14564 floats = 58256 B

// Base decomposition low-pass filters (pywt conventions from the reference).
__constant__ float cDecLo[4][12] = {
  {0.7071067811865476f, 0.7071067811865476f, 0,0,0,0,0,0,0,0,0,0},
  {-0.010597401784997278f, 0.032883011666982945f, 0.030841381835986965f, -0.18703481171888114f,
   -0.02798376941698385f, 0.6308807679295904f, 0.7148465705525415f, 0.23037781330885523f, 0,0,0,0},
  {-0.007800708325034148f, 0.0017677118642428036f, 0.04472490177066578f, -0.021060292512300564f,
   -0.07263752278646252f, 0.3379294217276218f, 0.787641141030194f, 0.4910559419267466f,
   -0.048311742585633f, -0.11799011114819057f, 0.0034907120842174702f, 0.015404109327027373f},
  {-0.08838834764831845f, 0.08838834764831845f, 0.7071067811865476f, 0.7071067811865476f,
    0.08838834764831845f, -0.08838834764831845f, 0,0,0,0,0,0}
};
__constant__ float cBiorDecHi[6] = {0,0,-0.7071067811865476f,0.7071067811865476f,0,0};
__constant__ float cBiorRecLo[6] = {0,0,0.7071067811865476f,0.7071067811865476f,0,0};
__constant__ float cBiorRecHi[6] = {-0.08838834764831845f,-0.08838834764831845f,0.7071067811865476f,
                                    -0.7071067811865476f,0.08838834764831845f,0.08838834764831845f};
__constant__ int cLn[4] = {2, 8, 12, 6};
__constant__ int cP[4]  = {0, 6, 10, 4};   // (2L-3)//2

__device__ __forceinline__ float dec_val(int wv, int hi, int k) {
  const int L = cLn[wv];
  if (k < 0 || k >= L) return 0.f;
  if (wv == 3) return hi ? cBiorDecHi[k] : cDecLo[3][k];
  if (!hi) return cDecLo[wv][k];
  const float v = cDecLo[wv][L - 1 - k];     // dec_hi[k] = (-1)^(k+1) dec_lo[L-1-k]
  return (k & 1) ? v : -v;
}

__device__ __forceinline__ float rec_val(int wv, int hi, int u) {
  const int L = cLn[wv];
  if (u < 0 || u >= L) return 0.f;
  if (wv == 3) return hi ? cBiorRecHi[u] : cBiorRecLo[u];
  if (!hi) return cDecLo[wv][L - 1 - u];     // rec_lo = reverse(dec_lo)
  const float v = cDecLo[wv][u];             // rec_hi[u] = (-1)^u dec_lo[u]
  return (u & 1) ? -v : v;
}

__device__ __forceinline__ v8f wmma4(v2f a, v2f b, v8f c) {
  // V_WMMA_F32_16X16X4_F32 : D(16x16,f32) = A(16x4,f32) x B(4x16,f32) + C
  return __builtin_amdgcn_wmma_f32_16x16x4_f32(
      false, a, false, b, (short)0, c, false, false);
}

// One TDM descriptor load: 2D tile, 16 f32 per row, kXUSE rows, row stride kE f32,
// LDS padding of 1 DWORD every 16 DWORDs -> matches xs[][kXSTR] layout.
__device__ __forceinline__ void tdm_load_tile(const float* gsrc) {
  const uint64_t ga = (uint64_t)(uintptr_t)gsrc;
  u32x4 g0;
  g0[0] = 1u;                                      // count=1, user mode
  g0[1] = 0u;                                      // lds_addr: xs at base of dyn LDS
  g0[2] = (unsigned)(ga & 0xFFFFFFFFu);            // global_addr[31:0]
  g0[3] = (unsigned)((ga >> 32) & 0x01FFFFFFu) | (2u << 30);  // addr[56:32] | type=2
  i32x8 g1;
  g1[0] = (2 << 16) | (1 << 20) | (3 << 22);       // data_size=4B, pad_en, interval=16dw (amount=1dw)
  g1[1] = (int)((unsigned)(kE & 0xFFFF) << 16);    // tensor_dim0 = 1024 (lo16)
  g1[2] = (int)(((unsigned)(kB * kS) & 0xFFFF) << 16); // dim0 hi16=0 | tensor_dim1 lo16
  g1[3] = (int)(16u << 16);                        // tensor_dim1 hi16=0 | tile_dim0=16
  g1[4] = kXUSE;                                   // tile_dim1 = 174, tile_dim2 = 0
  g1[5] = kE;                                      // tensor_dim0_stride lo32 = 1024
  g1[6] = 0;                                       // stride hi | tensor_dim1_stride lo
  g1[7] = 0;
  i32x4 g2 = {0, 0, 0, 0};
  i32x4 g3 = {0, 0, 0, 0};
#if __clang_major__ >= 23
  i32x8 g4 = {0, 0, 0, 0, 0, 0, 0, 0};
  __builtin_amdgcn_tensor_load_to_lds(g0, g1, g2, g3, g4, 0);
#else
  __builtin_amdgcn_tensor_load_to_lds(g0, g1, g2, g3, 0);
#endif
}

extern "C" __global__ __launch_bounds__(256)
void wlm_stage1(const float* __restrict__ x, const float* __restrict__ lamA,
                const float* __restrict__ lamD, const float* __restrict__ lgam,
                const float* __restrict__ th, const float* __restrict__ logits,
                float* __restrict__ y)
{
  extern __shared__ float smem[];
  float* xs   = smem + kOffXS;    // [kXWIN][17]   staged x tile
  float* cf   = smem + kOffCF;    // [(m*8+j)*17+c] shrunk coeffs
  float* Ftab = smem + kOffFT;    // [16 taps][16 subband cols]
  float* Gpad = smem + kOffGP;    // [8][48] zero-padded rec filters (idx v+16)
  float* prm  = smem + kOffPRM;   // lamA[16] lamD[16] (gamma*cos)[16]
  float* wtab = smem + kOffWT;    // softmax basis weights [4]

  const int tid = threadIdx.x;
  const int e0  = blockIdx.x * 16;
  const int so0 = blockIdx.y * kTS;
  const int b   = blockIdx.z;

  // ---- stage x tile (+halo) into LDS ----
  const bool interior = (so0 >= kXOFF) && (so0 - kXOFF + kXUSE <= kS);
  if (interior) {
    // Tensor Data Mover: one descriptor, issued by wave 0 only (TDM ignores EXEC,
    // so gate with a scalar-provable uniform branch).
    if ((__builtin_amdgcn_readfirstlane(tid) >> 5) == 0) {
      const float* gsrc = x + ((size_t)(b * kS + so0 - kXOFF) * kE + e0);
      tdm_load_tile(gsrc);
      __builtin_amdgcn_s_wait_tensorcnt((short)0);
    }
  } else {
    // boundary tiles: manual loads with reflect padding
    const int ch = tid & 15;
    for (int t = tid >> 4; t < kXUSE; t += 16) {
      int s = so0 - kXOFF + t;
      s = (s < 0) ? -s : s;
      s = (s >= kS) ? (2 * kS - 2 - s) : s;
      xs[t * kXSTR + ch] = x[((size_t)b * kS + s) * kE + e0 + ch];
    }
  }
  // warm L2 with the head of the next s-tile (global_prefetch_b8)
  if (so0 + kTS < kS) {
    __builtin_prefetch(&x[((size_t)b * kS + so0 + kTS - kXOFF + (tid >> 4)) * kE + e0 + (tid & 15)], 0, 1);
  }

  // ---- build unified filter tables + per-channel params ----
  {
    const int w = tid >> 4, j = tid & 15;
    float fv = 0.f;
    if (j < 8) {
      const int wv = j >> 1, hi = j & 1;
      const int idx = w - (10 - cP[wv]);          // alignment shift to POFF=10
      fv = dec_val(wv, hi, cLn[wv] - 1 - idx);    // filt = dec[::-1]
    }
    Ftab[w * 16 + j] = fv;
    for (int idx = tid; idx < 8 * 48; idx += 256) {
      const int jj = idx / 48, v16 = idx % 48, v = v16 - 16;
      float gv = 0.f;
      if (v >= 0 && v < 16)
        gv = rec_val(jj >> 1, jj & 1, v - (10 - cP[jj >> 1]));
      Gpad[idx] = gv;
    }
    if (tid < 16) {
      const int e = e0 + tid;
      prm[tid]      = lamA[e];
      prm[16 + tid] = lamD[e];
      prm[32 + tid] = (log1pf(expf(lgam[e])) + 1e-6f) * cosf(th[e]);  // gamma*cos(theta)
    }
    if (tid < 4) {   // softmax basis weights, computed once
      const float l0 = logits[0], l1 = logits[1], l2 = logits[2], l3 = logits[3];
      const float mx = fmaxf(fmaxf(l0, l1), fmaxf(l2, l3));
      const float e0w = expf(l0 - mx), e1w = expf(l1 - mx),
                  e2w = expf(l2 - mx), e3w = expf(l3 - mx);
      const float inv = 1.f / (e0w + e1w + e2w + e3w);
      wtab[tid] = (tid == 0 ? e0w : tid == 1 ? e1w : tid == 2 ? e2w : e3w) * inv;
    }
  }

  __syncthreads();

  const int lane = tid & 31;
  const int wave = tid >> 5;
  const int li   = lane & 15;   // A row (time) / B col (subband or channel)
  const int kk   = lane >> 4;   // K half-select

  // ---- DWT: 5 blocks x 16 coeff positions, all 8 subbands at once per channel ----
  for (int blk = wave; blk < 5; blk += 8) {
    const int m0 = -4 + blk * 16;
    // B (filter taps) is invariant across channels: hoist
    v2f Bg[4];
#pragma unroll
    for (int g = 0; g < 4; ++g) {
      const int k0 = 4 * g + 2 * kk;
      Bg[g].x = Ftab[(k0 + 0) * 16 + li];
      Bg[g].y = Ftab[(k0 + 1) * 16 + li];
    }
    for (int c = 0; c < 16; ++c) {
      v8f acc = {0, 0, 0, 0, 0, 0, 0, 0};
#pragma unroll
      for (int g = 0; g < 4; ++g) {
        const int tb = 2 * (m0 + li) + 4 * g + 2 * kk - 10 + kXOFF;
        v2f A;
        A.x = xs[tb * kXSTR + c];
        A.y = xs[(tb + 1) * kXSTR + c];
        acc = wmma4(A, Bg[g], acc);
      }
      // spectral shrinkage * softmax weight, coefficients -> LDS
      if (li < 8) {
        const float lam = (li & 1) ? prm[16 + c] : prm[c];
        const float scale = prm[32 + c] * wtab[li >> 1];
#pragma unroll
        for (int r = 0; r < 8; ++r) {
          const float z = acc[r];
          float t = fabsf(z) - lam;
          t = t > 0.f ? t : 0.f;
          const int m = m0 + r + 8 * kk + kCFOFF;
          cf[(m * 8 + li) * kXSTR + c] = copysignf(t, z) * scale;
        }
      }
    }
  }

  __syncthreads();

  // ---- IDWT: constant reconstruction matrix x 16-channel coeff block ----
  for (int blk = wave; blk < 8; blk += 8) {
    const int n0 = blk * 16;
    const int m0 = n0 / 2 - 3;          // chosen so v = i - 2k + 16 in [0,16)
    v8f acc = {0, 0, 0, 0, 0, 0, 0, 0};
    for (int j = 0; j < 8; ++j) {
#pragma unroll
      for (int g = 0; g < 4; ++g) {
        const int kabs = 4 * g + 2 * kk;
        v2f A, Bv;
        A.x = Gpad[j * 48 + (li - 2 * kabs + 32)];   // zero-padded: no selects
        A.y = Gpad[j * 48 + (li - 2 * kabs + 30)];
        const int mB = m0 + kabs + kCFOFF;
        Bv.x = cf[(mB * 8 + j) * kXSTR + li];
        Bv.y = cf[((mB + 1) * 8 + j) * kXSTR + li];
        acc = wmma4(A, Bv, acc);
      }
    }
    // residual add + store (D cols = channels -> contiguous 64B segments)
#pragma unroll
    for (int r = 0; r < 8; ++r) {
      const int trel = n0 + r + 8 * kk;
      const float yv = acc[r] + xs[(trel + kXOFF) * kXSTR + li];
      y[((size_t)b * kS + so0 + trel) * kE + e0 + li] = yv;
    }
  }
}

extern "C" __global__ __launch_bounds__(256)
void wlm_ln(float* __restrict__ y, const float* __restrict__ lnw, const float* __restrict__ lnb)
{
  __shared__ float s1[256];
  __shared__ float s2[256];
  const int tid = threadIdx.x;
  const size_t base = (size_t)blockIdx.x * kE;
  const float4 v = *reinterpret_cast<const float4*>(y + base + tid * 4);
  s1[tid] = v.x + v.y + v.z + v.w;
  s2[tid] = v.x * v.x + v.y * v.y + v.z * v.z + v.w * v.w;
  __syncthreads();
  for (int o = 128; o > 0; o >>= 1) {
    if (tid < o) { s1[tid] += s1[tid + o]; s2[tid] += s2[tid + o]; }
    __syncthreads();
  }
  const float mean = s1[0] * (1.f / kE);
  const float var  = s2[0] * (1.f / kE) - mean * mean;
  const float rstd = rsqrtf(var + 1e-5f);
  const float4 w  = *reinterpret_cast<const float4*>(lnw + tid * 4);
  const float4 bb = *reinterpret_cast<const float4*>(lnb + tid * 4);
  float4 o4;
  o4.x = (v.x - mean) * rstd * w.x + bb.x;
  o4.y = (v.y - mean) * rstd * w.y + bb.y;
  o4.z = (v.z - mean) * rstd * w.z + bb.z;
  o4.w = (v.w - mean) * rstd * w.w + bb.w;
  *reinterpret_cast<float4*>(y + base + tid * 4) = o4;
}

extern "C" void kernel_launch(void* const* d_in, const int* in_sizes, int n_in,
                              void* d_out, int out_size, void* d_ws, size_t ws_size,
                              hipStream_t stream) {
  (void)in_sizes; (void)n_in; (void)out_size; (void)d_ws; (void)ws_size;
  const float* x     = (const float*)d_in[0];
  const float* lamA  = (const float*)d_in[1];
  const float* lamD  = (const float*)d_in[2];
  const float* lgam  = (const float*)d_in[3];
  const float* th    = (const float*)d_in[4];
  const float* logit = (const float*)d_in[5];
  const float* lnw   = (const float*)d_in[6];
  const float* lnb   = (const float*)d_in[7];
  float* y = (float*)d_out;

  const size_t smem = (size_t)kSmemF * sizeof(float);
  dim3 g1(kE / 16, kS / kTS, kB);
  wlm_stage1<<<g1, 256, smem, stream>>>(x, lamA, lamD, lgam, th, logit, y);
  wlm_ln<<<dim3(kB * kS), 256, 0, stream>>>(y, lnw, lnb);
}